// SAGE_50276887167532
// MI455X (gfx1250) — compile-verified
//
#include <hip/hip_runtime.h>

#define N_NODES 100000
#define N_EDGES 1200000
#define D 64
#define M_TILES 6250          // 100000 / 16 exactly
#define GEMM_BLOCKS 3125      // 6250*4 tiles / 8 waves per block

typedef __attribute__((ext_vector_type(2))) float v2f;
typedef __attribute__((ext_vector_type(8))) float v8f;

// ---------------- utility kernels ----------------

__global__ void zero_f32(float* __restrict__ p, int n) {
  int i = blockIdx.x * blockDim.x + threadIdx.x;
  if (i < n) p[i] = 0.0f;
}

__global__ void degree_kernel(const long long* __restrict__ ei,
                              float* __restrict__ deg) {
  int e = blockIdx.x * blockDim.x + threadIdx.x;
  if (e < N_EDGES) {
    long long d = ei[(size_t)N_EDGES + e];
    atomicAdd(&deg[d], 1.0f);
  }
}

__global__ void invdeg_kernel(float* __restrict__ deg) {
  int i = blockIdx.x * blockDim.x + threadIdx.x;
  if (i < N_NODES) deg[i] = 1.0f / fmaxf(deg[i], 1.0f);
}

// One wave32 per edge: each lane moves 2 consecutive floats (64 feats / 32 lanes).
__global__ void scatter_kernel(const long long* __restrict__ ei,
                               const float* __restrict__ x,
                               float* __restrict__ agg) {
  int warp = threadIdx.x >> 5;
  int lane = threadIdx.x & 31;
  int e = blockIdx.x * 8 + warp;
  if (e >= N_EDGES) return;
  long long s = ei[e];
  long long d = ei[(size_t)N_EDGES + e];
  float2 v = ((const float2*)x)[(size_t)s * 32 + lane];
  float* dst = agg + (size_t)d * 64 + lane * 2;
  atomicAdd(dst,     v.x);
  atomicAdd(dst + 1, v.y);
}

// ---------------- WMMA GEMM: out[M,64] = f(A[M,64] @ W[64,64]) ----------------
// Each wave computes one 16x16 tile of the output. 8 waves / 256-thread block.
// K=64 consumed in 16 steps of V_WMMA_F32_16X16X4_F32 (exact fp32, matches ref).
template <bool SCALE, bool EPILOGUE, bool RELU>
__global__ void gemm64_kernel(const float* __restrict__ A,
                              const float* __restrict__ W,
                              const float* __restrict__ invdeg,
                              const float* __restrict__ bias,
                              const float* __restrict__ addend,
                              float* __restrict__ out) {
  __shared__ float ldsW[64 * 64];  // 16 KB of the 320 KB/WGP LDS
  {
    // cooperative 128-bit loads: 4096 floats = 1024 float4, 256 threads
    const float4* Wv = (const float4*)W;
    float4* Lv = (float4*)ldsW;
    for (int i = threadIdx.x; i < 1024; i += 256) Lv[i] = Wv[i];
  }
  __syncthreads();

  int warp = threadIdx.x >> 5;
  int lane = threadIdx.x & 31;
  int tile = blockIdx.x * 8 + warp;   // 25000 tiles total, exact fill
  int nt = tile & 3;                  // 4 N-tiles (64/16)
  int mt = tile >> 2;                 // 6250 M-tiles

  int half = lane >> 4;               // 0: lanes 0-15, 1: lanes 16-31
  int l15  = lane & 15;
  int m    = mt * 16 + l15;           // A-fragment row (both halves span M=0..15)
  int n    = nt * 16 + l15;           // B/C/D column
  int koff = half * 2;                // A: K={0,1} vs {2,3}; B mirrors

  float scale = 1.0f;
  if (SCALE) scale = invdeg[m];

  const float* Arow = A + (size_t)m * 64;
  v8f c = {};
#pragma unroll
  for (int k0 = 0; k0 < 64; k0 += 4) {
    v2f a, b;
    a.x = Arow[k0 + koff];
    a.y = Arow[k0 + koff + 1];
    if (SCALE) { a.x *= scale; a.y *= scale; }
    b.x = ldsW[(k0 + koff) * 64 + n];
    b.y = ldsW[(k0 + koff + 1) * 64 + n];
    // 8-arg f32 pattern: (neg_a, A, neg_b, B, c_mod, C, reuse_a, reuse_b)
    c = __builtin_amdgcn_wmma_f32_16x16x4_f32(false, a, false, b,
                                              (short)0, c, false, false);
  }

  // C/D layout: lanes 0-15 hold M=v, lanes 16-31 hold M=v+8 (v = VGPR 0..7)
  int mbase = mt * 16 + half * 8;
  float bn = 0.0f;
  if (EPILOGUE) bn = bias[n];
#pragma unroll
  for (int v = 0; v < 8; ++v) {
    float val = c[v];
    if (EPILOGUE) val += bn + addend[(size_t)(mbase + v) * 64 + n];
    if (RELU) val = fmaxf(val, 0.0f);
    out[(size_t)(mbase + v) * 64 + n] = val;
  }
}

// ---------------- launch ----------------

extern "C" void kernel_launch(void* const* d_in, const int* in_sizes, int n_in,
                              void* d_out, int out_size, void* d_ws, size_t ws_size,
                              hipStream_t stream) {
  (void)in_sizes; (void)n_in; (void)out_size; (void)ws_size;

  const float*     x   = (const float*)d_in[0];
  const long long* ei  = (const long long*)d_in[1];  // int64 edge_index [2, E]
  const float*     Wl0 = (const float*)d_in[2];
  const float*     bl0 = (const float*)d_in[3];
  const float*     Wr0 = (const float*)d_in[4];
  const float*     Wl1 = (const float*)d_in[5];
  const float*     bl1 = (const float*)d_in[6];
  const float*     Wr1 = (const float*)d_in[7];
  float* out = (float*)d_out;

  // workspace: agg | xr | h0 | invdeg  (~77.2 MB)
  float* agg = (float*)d_ws;
  float* xr  = agg + (size_t)N_NODES * D;
  float* h0  = xr  + (size_t)N_NODES * D;
  float* idg = h0  + (size_t)N_NODES * D;

  const int feat_elems = N_NODES * D;
  dim3 b256(256);

  // degree -> 1/max(deg,1)
  zero_f32<<<(N_NODES + 255) / 256, b256, 0, stream>>>(idg, N_NODES);
  degree_kernel<<<(N_EDGES + 255) / 256, b256, 0, stream>>>(ei, idg);
  invdeg_kernel<<<(N_NODES + 255) / 256, b256, 0, stream>>>(idg);

  // ---- layer 0 ----
  zero_f32<<<(feat_elems + 255) / 256, b256, 0, stream>>>(agg, feat_elems);
  scatter_kernel<<<N_EDGES / 8, b256, 0, stream>>>(ei, x, agg);
  gemm64_kernel<false, false, false><<<GEMM_BLOCKS, b256, 0, stream>>>(
      x, Wr0, nullptr, nullptr, nullptr, xr);
  gemm64_kernel<true, true, true><<<GEMM_BLOCKS, b256, 0, stream>>>(
      agg, Wl0, idg, bl0, xr, h0);

  // ---- layer 1 ----
  zero_f32<<<(feat_elems + 255) / 256, b256, 0, stream>>>(agg, feat_elems);
  scatter_kernel<<<N_EDGES / 8, b256, 0, stream>>>(ei, h0, agg);
  gemm64_kernel<false, false, false><<<GEMM_BLOCKS, b256, 0, stream>>>(
      h0, Wr1, nullptr, nullptr, nullptr, xr);
  gemm64_kernel<true, true, false><<<GEMM_BLOCKS, b256, 0, stream>>>(
      agg, Wl1, idg, bl1, xr, out);
}